// LogSig_v1_1614907703721
// MI455X (gfx1250) — compile-verified
//
#include <hip/hip_runtime.h>

typedef float v2f __attribute__((ext_vector_type(2)));
typedef float v8f __attribute__((ext_vector_type(8)));
typedef int   v4i __attribute__((vector_size(4 * sizeof(int))));

typedef __attribute__((address_space(1))) v4i* g_v4i_p;   // global (device) 16B vector ptr
typedef __attribute__((address_space(3))) v4i* l_v4i_p;   // LDS 16B vector ptr

#define BB   32
#define TT   256
#define DD   64
#define NSEG 32
#define PITCH 68                          // row stride in floats; 272B, 16B-aligned
#define OUTF (DD + (DD * (DD - 1)) / 2)   // 64 + 2016 = 2080 features per (b,s)

// starts = int(round(linspace(1, T, NSEG+1))) - 1, with Python banker's rounding.
// v = 1 + s*255/32 = (32 + 255*s)/32, exact; round-half-to-even in integers.
__device__ __forceinline__ int seg_start(int s) {
    int num = 32 + s * 255;
    int q = num >> 5;
    int r = num & 31;
    if (r > 16)       q += 1;
    else if (r == 16) q += (q & 1);   // ties to even
    return q - 1;
}

__global__ __launch_bounds__(512) void logsig2_wmma(const float* __restrict__ inp,
                                                    float* __restrict__ out) {
    __shared__ float Xs[17][PITCH];    // segment rows x_t (16B-aligned rows)
    __shared__ float DXs[16][PITCH];   // dx_t = x_{t+1} - x_t, zero-padded in K
    __shared__ float Msh[64][65];      // Mseg = X^T * DX (64x64 f32)

    const int s   = blockIdx.x;
    const int b   = blockIdx.y;
    const int tid = threadIdx.x;
    const int t0  = seg_start(s);
    const int t1  = seg_start(s + 1);
    const int nt  = t1 - t0;                // segment length (7 or 8)
    const int ntk = (nt + 3) & ~3;          // K padded to multiple of 4

    const float* src = inp + ((size_t)b * TT + t0) * DD;

    // ---- Stage rows t0..t1 ((nt+1) x 64 floats) into LDS ----
#if defined(__gfx1250__) && __has_builtin(__builtin_amdgcn_global_load_async_to_lds_b128)
    {
        // One 16B async copy per thread; rows are 256B in global, 272B in LDS.
        const int nq = (nt + 1) * (DD / 4);     // float4 quanta (<= 144)
        for (int q4 = tid; q4 < nq; q4 += 512) {
            int r = q4 >> 4, c4 = (q4 & 15) << 2;
            __builtin_amdgcn_global_load_async_to_lds_b128(
                (g_v4i_p)(src + q4 * 4),
                (l_v4i_p)&Xs[r][c4],
                /*offset=*/0, /*cpol=*/0);
        }
#if __has_builtin(__builtin_amdgcn_s_wait_asynccnt)
        __builtin_amdgcn_s_wait_asynccnt(0);
#else
        asm volatile("s_wait_asynccnt 0" ::: "memory");
#endif
    }
#else
    __builtin_prefetch(src + tid, 0, 1);    // global_prefetch_b8
    for (int i = tid; i < (nt + 1) * DD; i += 512)
        Xs[i >> 6][i & 63] = src[i];
#endif
    // Defensive zero of unused rows (K padding region for the A operand).
    for (int i = (nt + 1) * DD + tid; i < 17 * DD; i += 512)
        Xs[i >> 6][i & 63] = 0.0f;
    __syncthreads();

    // DX rows, zero-padded up to ntk so padded K contributes exactly 0.
    for (int i = tid; i < ntk * DD; i += 512) {
        int r = i >> 6, c = i & 63;
        DXs[r][c] = (r < nt) ? (Xs[r + 1][c] - Xs[r][c]) : 0.0f;
    }
    __syncthreads();

    // 16 waves: wave (di,ei) computes the 16x16 tile Mseg[di.., ei..]
    // as a chain of V_WMMA_F32_16X16X4_F32 over the (padded) K dimension.
    const int wave = tid >> 5;
    const int lane = tid & 31;
    const int hl   = lane >> 4;      // half-wave selector
    const int lm   = lane & 15;
    const int di   = (wave >> 2) * 16;
    const int ei   = (wave & 3) * 16;

    v8f acc = {0.f, 0.f, 0.f, 0.f, 0.f, 0.f, 0.f, 0.f};
    for (int c = 0; c < ntk; c += 4) {
        // A 16x4 (M=d, K=t): lane<16 -> K = c+0,c+1 ; lane>=16 -> K = c+2,c+3
        // B 4x16 (K=t, N=e): same half-wave K striping, N = lane%16
        v2f av, bv;
        av.x = Xs[c + 2 * hl + 0][di + lm];
        av.y = Xs[c + 2 * hl + 1][di + lm];
        bv.x = DXs[c + 2 * hl + 0][ei + lm];
        bv.y = DXs[c + 2 * hl + 1][ei + lm];
        acc = __builtin_amdgcn_wmma_f32_16x16x4_f32(
            /*neg_a=*/false, av, /*neg_b=*/false, bv,
            /*c_mod=*/(short)0, acc, /*reuse_a=*/false, /*reuse_b=*/false);
    }

    // C/D layout: VGPR r -> row r (lanes 0-15) / r+8 (lanes 16-31), col = lane%16
#pragma unroll
    for (int r = 0; r < 8; ++r)
        Msh[di + r + 8 * hl][ei + lm] = acc[r];
    __syncthreads();

    float* op = out + ((size_t)b * NSEG + s) * OUTF;

    // s1 = xb - xa  (first 64 outputs)
    if (tid < DD)
        op[tid] = Xs[nt][tid] - Xs[0][tid];

    // A = 0.5*(t2 - t2^T), t2 = Mseg - xa (x) s1 ; emit strict upper triangle
    // A[i][j] = 0.5*((Mseg[i][j]-Mseg[j][i]) - (xa_i*s1_j - xa_j*s1_i))
    for (int q = tid; q < DD * DD; q += 512) {
        int i = q >> 6, j = q & 63;
        if (i < j) {
            float xai = Xs[0][i], xaj = Xs[0][j];
            float s1i = Xs[nt][i] - xai;
            float s1j = Xs[nt][j] - xaj;
            float val = 0.5f * ((Msh[i][j] - Msh[j][i]) - (xai * s1j - xaj * s1i));
            int p = i * 63 - (i * (i - 1)) / 2 + (j - i - 1);   // tri-upper linear idx
            op[DD + p] = val;
        }
    }
}

extern "C" void kernel_launch(void* const* d_in, const int* in_sizes, int n_in,
                              void* d_out, int out_size, void* d_ws, size_t ws_size,
                              hipStream_t stream) {
    const float* inp = (const float*)d_in[0];
    float* out = (float*)d_out;
    (void)in_sizes; (void)n_in; (void)out_size; (void)d_ws; (void)ws_size;
    dim3 grid(NSEG, BB);   // one block per (segment, batch)
    logsig2_wmma<<<grid, 512, 0, stream>>>(inp, out);
}